// PSNRLoss_78623671320782
// MI455X (gfx1250) — compile-verified
//
#include <hip/hip_runtime.h>
#include <hip/hip_bf16.h>

// Problem constants (from reference setup_inputs)
#define N_ITERS 12
#define N_BATCH 4
#define N_CH    3
#define IMG_H   384
#define IMG_W   512
#define HW      (IMG_H * IMG_W)          // 196608
#define TOTAL_PIX (N_BATCH * HW)         // 786432
#define PIX_PER_THREAD 4
#define BLOCK 256
#define BLOCKS_X (TOTAL_PIX / (BLOCK * PIX_PER_THREAD))   // 768
#define CASCADE 0.85f

typedef float v2f __attribute__((ext_vector_type(2)));
typedef float v4f __attribute__((ext_vector_type(4)));
typedef float v8f __attribute__((ext_vector_type(8)));

// ---------------------------------------------------------------------------
// Per-pixel bilinear warp + 3-channel squared error (zeros padding, as ref).
// ---------------------------------------------------------------------------
__device__ __forceinline__ float pixel_se3(const float* __restrict__ f1b,
                                           const v4f* __restrict__ t2,
                                           int j, int x, int y,
                                           float fx, float fy) {
    const float px  = (float)x + fx;
    const float py  = (float)y + fy;
    const float x0f = floorf(px);
    const float y0f = floorf(py);
    const int   x0  = (int)x0f;
    const int   y0  = (int)y0f;
    const float wx1 = px - x0f, wx0 = 1.0f - wx1;
    const float wy1 = py - y0f, wy0 = 1.0f - wy1;

    const bool vx0 = (x0 >= 0)  && (x0 <  IMG_W);
    const bool vx1 = (x0 >= -1) && (x0 <  IMG_W - 1);
    const bool vy0 = (y0 >= 0)  && (y0 <  IMG_H);
    const bool vy1 = (y0 >= -1) && (y0 <  IMG_H - 1);
    const int  xc0 = min(max(x0,     0), IMG_W - 1);
    const int  xc1 = min(max(x0 + 1, 0), IMG_W - 1);
    const int  yc0 = min(max(y0,     0), IMG_H - 1);
    const int  yc1 = min(max(y0 + 1, 0), IMG_H - 1);

    const float w00 = wy0 * wx0, w01 = wy0 * wx1;
    const float w10 = wy1 * wx0, w11 = wy1 * wx1;

    float se = 0.0f;
    #pragma unroll
    for (int c = 0; c < N_CH; ++c) {
        const float* __restrict__ img = f1b + (size_t)c * HW;
        const float v00 = (vy0 && vx0) ? img[yc0 * IMG_W + xc0] : 0.0f;
        const float v01 = (vy0 && vx1) ? img[yc0 * IMG_W + xc1] : 0.0f;
        const float v10 = (vy1 && vx0) ? img[yc1 * IMG_W + xc0] : 0.0f;
        const float v11 = (vy1 && vx1) ? img[yc1 * IMG_W + xc1] : 0.0f;
        const float est = v00 * w00 + v01 * w01 + v10 * w10 + v11 * w11;
        const float e   = t2[c][j] - est;
        se = fmaf(e, e, se);
    }
    return se;
}

// ---------------------------------------------------------------------------
// Kernel 1: fused warp + SE, 4 pixels/thread with B128 streaming loads.
//   grid = (768, 12), block = 256.  One deterministic partial per block.
//   Wave reduction via V_WMMA_F32_16X16X4_F32 + ds_swizzle SWAPX16.
// ---------------------------------------------------------------------------
__global__ __launch_bounds__(256) void psnr_warp_se_kernel(
    const float* __restrict__ flow,      // [12,4,2,H,W], c0=dy, c1=dx
    const float* __restrict__ f1,        // [4,3,H,W]
    const float* __restrict__ f2,        // [4,3,H,W]
    float* __restrict__ partials)        // [12, 768]
{
    const int it  = blockIdx.y;
    const int gid = blockIdx.x * BLOCK + threadIdx.x;   // 0 .. TOTAL_PIX/4-1
    const int p4  = gid << 2;                           // first pixel of group
    const int b   = p4 / HW;
    const int p   = p4 - b * HW;
    const int y   = p >> 9;                // p / 512
    const int x   = p & (IMG_W - 1);       // p % 512; group stays in one row

    // Streaming loads: 16B-aligned GLOBAL_LOAD_B128 (p4 % 4 == 0).
    const size_t fbase = ((size_t)(it * N_BATCH + b) * 2) * (size_t)HW + (size_t)p;
    const v4f fy4 = *(const v4f*)(flow + fbase);             // dy plane
    const v4f fx4 = *(const v4f*)(flow + fbase + (size_t)HW); // dx plane

    const float* __restrict__ f1b = f1 + (size_t)b * N_CH * HW;
    const float* __restrict__ f2b = f2 + (size_t)b * N_CH * HW;
    v4f t2[N_CH];
    #pragma unroll
    for (int c = 0; c < N_CH; ++c)
        t2[c] = *(const v4f*)(f2b + (size_t)c * HW + (size_t)p);

    // Two accumulators per lane -> fills the 16x4 f32 A-matrix (2 VGPRs/lane).
    float acc0 = 0.0f, acc1 = 0.0f;
    #pragma unroll
    for (int j = 0; j < PIX_PER_THREAD; ++j) {
        const float se3 = pixel_se3(f1b, t2, j, x + j, y, fx4[j], fy4[j]);
        if (j & 1) acc1 += se3; else acc0 += se3;
    }

    // ---- Wave-level sum of (acc0, acc1) across all 32 lanes ----
    // A (16x4 f32): lane L<16 holds A[L,0..1]; lane L>=16 holds A[L-16,2..3].
    // B = ones(4x16)  =>  D[m,n] = rowsum(A row m) for every n.
    // Lane's 8 D regs sum to rows0-7 (lanes 0-15) / rows8-15 (lanes 16-31);
    // SWAPX16 swizzle merges the halves -> full 64-value wave sum, exact f32.
    float wavesum;
#if __has_builtin(__builtin_amdgcn_wmma_f32_16x16x4_f32)
    v2f a;    a[0] = acc0;  a[1] = acc1;
    v2f ones; ones[0] = 1.0f; ones[1] = 1.0f;
    v8f cz = {};
    v8f d = __builtin_amdgcn_wmma_f32_16x16x4_f32(
        /*neg_a=*/false, a, /*neg_b=*/false, ones,
        /*c_mod=*/(short)0, cz, /*reuse_a=*/false, /*reuse_b=*/false);
    float part = d[0] + d[1] + d[2] + d[3] + d[4] + d[5] + d[6] + d[7];
  #if __has_builtin(__builtin_amdgcn_ds_swizzle)
    const int sw = __builtin_amdgcn_ds_swizzle(
        __builtin_bit_cast(int, part), 0x401F);   // xor 0x10: lane ^= 16
    wavesum = part + __builtin_bit_cast(float, sw);
  #else
    wavesum = part + __shfl_xor(part, 16, 32);
  #endif
#else
    float part = acc0 + acc1;
    #pragma unroll
    for (int off = 16; off > 0; off >>= 1) part += __shfl_xor(part, off, 32);
    wavesum = part;
#endif

    // ---- Block-level reduction across 8 waves via LDS (fixed order) ----
    __shared__ float lsum[8];
    const int wave = threadIdx.x >> 5;
    const int lane = threadIdx.x & 31;
    if (lane == 0) lsum[wave] = wavesum;
    __syncthreads();
    if (threadIdx.x == 0) {
        float s = 0.0f;
        #pragma unroll
        for (int wv = 0; wv < 8; ++wv) s += lsum[wv];
        partials[it * BLOCKS_X + blockIdx.x] = s;   // deterministic, no atomics
    }
}

// ---------------------------------------------------------------------------
// Kernel 2: per-iteration deterministic tree-reduce of 768 partials, then
//           term_i = 10*log10(mse_i) * 0.85^(12-i)   (== -psnr_i * w_i)
// ---------------------------------------------------------------------------
__global__ __launch_bounds__(256) void psnr_reduce_kernel(
    const float* __restrict__ partials,  // [12, 768]
    float* __restrict__ terms)           // [12]
{
    const int it = blockIdx.x;
    const int t  = threadIdx.x;
    const float* __restrict__ p = partials + it * BLOCKS_X;

    __shared__ float l[BLOCK];
    l[t] = p[t] + p[t + 256] + p[t + 512];
    __syncthreads();
    #pragma unroll
    for (int s = 128; s > 0; s >>= 1) {
        if (t < s) l[t] += l[t + s];
        __syncthreads();
    }
    if (t == 0) {
        const float inv_n = 1.0f / (float)(N_BATCH * N_CH * HW);
        const float mse   = l[0] * inv_n;
        const float wgt   = powf(CASCADE, (float)(N_ITERS - it));
        terms[it] = 10.0f * log10f(mse) * wgt;
    }
}

// ---------------------------------------------------------------------------
// Kernel 3: loss = sum of the 12 terms
// ---------------------------------------------------------------------------
__global__ void psnr_finalize_kernel(const float* __restrict__ terms,
                                     float* __restrict__ out) {
    if (threadIdx.x == 0 && blockIdx.x == 0) {
        float loss = 0.0f;
        for (int i = 0; i < N_ITERS; ++i) loss += terms[i];
        out[0] = loss;
    }
}

// ---------------------------------------------------------------------------
extern "C" void kernel_launch(void* const* d_in, const int* in_sizes, int n_in,
                              void* d_out, int out_size, void* d_ws, size_t ws_size,
                              hipStream_t stream) {
    const float* flow = (const float*)d_in[0];   // [12,4,2,384,512]
    const float* f1   = (const float*)d_in[1];   // [4,3,384,512]
    const float* f2   = (const float*)d_in[2];   // [4,3,384,512]
    float* partials = (float*)d_ws;              // 12*768 floats
    float* terms    = partials + N_ITERS * BLOCKS_X;  // 12 floats
    float* out      = (float*)d_out;             // scalar loss

    dim3 grid(BLOCKS_X, N_ITERS);                // (768, 12): exact coverage
    psnr_warp_se_kernel<<<grid, BLOCK, 0, stream>>>(flow, f1, f2, partials);
    psnr_reduce_kernel<<<N_ITERS, BLOCK, 0, stream>>>(partials, terms);
    psnr_finalize_kernel<<<1, 32, 0, stream>>>(terms, out);
}